// STRNN_88545045775013
// MI455X (gfx1250) — compile-verified
//
#include <hip/hip_runtime.h>
#include <hip/hip_bf16.h>

typedef float v2f __attribute__((ext_vector_type(2)));
typedef float v8f __attribute__((ext_vector_type(8)));

#define Hd 512
#define Ld 512
#define Bd 64
#define LOC_STRIDE 516   // 516 % 64 == 4  -> conflict-free f32x2 A-frag reads
#define VT_STRIDE  500   // 500 % 64 == 52 -> conflict-free as well
#define EPS 1e-9f

// ---------------------------------------------------------------------------
// Zero the loc_vec accumulator in workspace.
// ---------------------------------------------------------------------------
__global__ __launch_bounds__(256) void strnn_zero_kernel(float* p, int n) {
    int i = blockIdx.x * 256 + threadIdx.x;
    if (i < n) p[i] = 0.0f;
}

// ---------------------------------------------------------------------------
// usr_vec[i] = sum_j weight_ih[i][j] * h0[j]   (tiny matvec, 512x512)
// ---------------------------------------------------------------------------
__global__ __launch_bounds__(256) void strnn_usr_kernel(const float* __restrict__ Wih,
                                                        const float* __restrict__ h0,
                                                        float* __restrict__ usr) {
    int i = blockIdx.x * 256 + threadIdx.x;
    if (i < Hd) {
        float s = 0.0f;
        #pragma unroll 4
        for (int j = 0; j < Hd; ++j) s = fmaf(Wih[i * Hd + j], h0[j], s);
        usr[i] = s;
    }
}

// ---------------------------------------------------------------------------
// Main fused kernel: one workgroup = 16 rows (one batch, 16 consecutive l).
// 128 threads = 4 waves; each wave owns 128 output columns (8 N-tiles).
// Stage 1: vt = a*(loc.Wtu^T) + b*(loc.Wtl^T)     (WMMA f32 16x16x4, K-loop)
// Stage 2: term = c*(vt.Wsu^T) + d*(vt.Wsl^T), masked row-sum -> atomicAdd.
// ---------------------------------------------------------------------------
__global__ __launch_bounds__(128) void strnn_main_kernel(
    const float* __restrict__ tdu, const float* __restrict__ tdl,
    const float* __restrict__ ldu, const float* __restrict__ ldl,
    const int*   __restrict__ cur_loc, const int* __restrict__ loc_len,
    const float* __restrict__ Wtu, const float* __restrict__ Wtl,
    const float* __restrict__ Wsu, const float* __restrict__ Wsl,
    const float* __restrict__ locw, float* __restrict__ loc_vec)
{
    __shared__ float lds_loc[16 * LOC_STRIDE];
    __shared__ float lds_vt [16 * VT_STRIDE];
    __shared__ float sa[16], sb[16], sc[16], sd[16], smk[16];
    __shared__ int   srow[16];

    const int b    = blockIdx.x >> 5;        // 32 tiles per batch
    const int tile = blockIdx.x & 31;
    const int l0   = tile * 16;
    const int tid  = threadIdx.x;

    // per-row mixing coefficients, mask, gather indices
    if (tid < 16) {
        const int idx = b * Ld + l0 + tid;
        float tu = tdu[idx], tl = tdl[idx];
        float lu = ldu[idx], ll = ldl[idx];
        float tden = tu + tl + EPS;
        float lden = lu + ll + EPS;
        sa[tid] = tu / tden;  sb[tid] = tl / tden;
        sc[tid] = lu / lden;  sd[tid] = ll / lden;
        smk[tid] = ((l0 + tid) < loc_len[b]) ? 1.0f : 0.0f;
        srow[tid] = cur_loc[idx];
    }
    __syncthreads();

    // gather 16 rows of location_weight into LDS (coalesced: 512B bursts/row)
    for (int i = tid; i < 16 * Hd; i += 128) {
        const int r = i >> 9;
        const int c = i & (Hd - 1);
        lds_loc[r * LOC_STRIDE + c] = locw[(size_t)srow[r] * Hd + c];
    }
    __syncthreads();

    const int lane = tid & 31;
    const int wave = tid >> 5;
    const int g    = lane >> 4;     // half-wave group (K/M split)
    const int hn   = lane & 15;     // A-frag row / B-frag col / C col

    // -------------------- Stage 1: vt tiles --------------------
    #pragma unroll 1
    for (int nt = 0; nt < 8; ++nt) {
        const int n0 = (wave * 8 + nt) * 16;
        v8f au = {}; v8f al = {};
        #pragma unroll 4
        for (int k = 0; k < Hd; k += 4) {
            v2f a, bu, bl;
            const float* lp = &lds_loc[hn * LOC_STRIDE + k + 2 * g];
            a.x = lp[0];  a.y = lp[1];
            const float* wu = Wtu + (size_t)(n0 + hn) * Hd + k + 2 * g;
            const float* wl = Wtl + (size_t)(n0 + hn) * Hd + k + 2 * g;
            bu.x = wu[0]; bu.y = wu[1];
            bl.x = wl[0]; bl.y = wl[1];
            au = __builtin_amdgcn_wmma_f32_16x16x4_f32(false, a, false, bu, (short)0, au, false, false);
            al = __builtin_amdgcn_wmma_f32_16x16x4_f32(false, a, false, bl, (short)0, al, false, false);
        }
        #pragma unroll
        for (int v = 0; v < 8; ++v) {
            const int row = v + 8 * g;
            lds_vt[row * VT_STRIDE + n0 + hn] = sa[row] * au[v] + sb[row] * al[v];
        }
    }
    __syncthreads();

    // -------------------- Stage 2: term tiles + masked row reduce --------------------
    #pragma unroll 1
    for (int nt = 0; nt < 8; ++nt) {
        const int n0 = (wave * 8 + nt) * 16;
        v8f au = {}; v8f al = {};
        #pragma unroll 4
        for (int k = 0; k < Hd; k += 4) {
            v2f a, bu, bl;
            const float* vp = &lds_vt[hn * VT_STRIDE + k + 2 * g];
            a.x = vp[0];  a.y = vp[1];
            const float* wu = Wsu + (size_t)(n0 + hn) * Hd + k + 2 * g;
            const float* wl = Wsl + (size_t)(n0 + hn) * Hd + k + 2 * g;
            bu.x = wu[0]; bu.y = wu[1];
            bl.x = wl[0]; bl.y = wl[1];
            au = __builtin_amdgcn_wmma_f32_16x16x4_f32(false, a, false, bu, (short)0, au, false, false);
            al = __builtin_amdgcn_wmma_f32_16x16x4_f32(false, a, false, bl, (short)0, al, false, false);
        }
        float part = 0.0f;
        #pragma unroll
        for (int v = 0; v < 8; ++v) {
            const int row = v + 8 * g;
            part += smk[row] * (sc[row] * au[v] + sd[row] * al[v]);
        }
        // fold the two half-wave row groups (rows 0-7 and 8-15) together
        part += __shfl_xor(part, 16, 32);
        if (g == 0) atomicAdd(&loc_vec[b * Hd + n0 + hn], part);
    }
}

// ---------------------------------------------------------------------------
// Softmax over H for each batch row: out[b] = softmax(loc_vec[b] + usr_vec)
// ---------------------------------------------------------------------------
__global__ __launch_bounds__(256) void strnn_softmax_kernel(const float* __restrict__ loc_vec,
                                                            const float* __restrict__ usr,
                                                            float* __restrict__ out) {
    __shared__ float red[256];
    const int b = blockIdx.x, tid = threadIdx.x;
    const float x0 = loc_vec[b * Hd + tid]       + usr[tid];
    const float x1 = loc_vec[b * Hd + 256 + tid] + usr[256 + tid];

    red[tid] = fmaxf(x0, x1);
    __syncthreads();
    for (int s = 128; s > 0; s >>= 1) {
        if (tid < s) red[tid] = fmaxf(red[tid], red[tid + s]);
        __syncthreads();
    }
    const float mx = red[0];
    __syncthreads();

    const float e0 = __expf(x0 - mx);
    const float e1 = __expf(x1 - mx);
    red[tid] = e0 + e1;
    __syncthreads();
    for (int s = 128; s > 0; s >>= 1) {
        if (tid < s) red[tid] += red[tid + s];
        __syncthreads();
    }
    const float inv = 1.0f / red[0];
    out[b * Hd + tid]       = e0 * inv;
    out[b * Hd + 256 + tid] = e1 * inv;
}

// ---------------------------------------------------------------------------
extern "C" void kernel_launch(void* const* d_in, const int* in_sizes, int n_in,
                              void* d_out, int out_size, void* d_ws, size_t ws_size,
                              hipStream_t stream) {
    const float* tdu     = (const float*)d_in[0];
    const float* tdl     = (const float*)d_in[1];
    const float* ldu     = (const float*)d_in[2];
    const float* ldl     = (const float*)d_in[3];
    const int*   cur_loc = (const int*)  d_in[4];
    const int*   loc_len = (const int*)  d_in[5];
    const float* Wih     = (const float*)d_in[6];
    const float* Wtu     = (const float*)d_in[7];
    const float* Wtl     = (const float*)d_in[8];
    const float* Wsu     = (const float*)d_in[9];
    const float* Wsl     = (const float*)d_in[10];
    const float* h0      = (const float*)d_in[11];
    const float* locw    = (const float*)d_in[12];
    float*       out     = (float*)d_out;

    float* ws_locvec = (float*)d_ws;                 // [B*H]
    float* ws_usr    = ws_locvec + Bd * Hd;          // [H]

    // 1) zero accumulator (+usr slot) each call — deterministic, graph-safe
    strnn_zero_kernel<<<(Bd * Hd + Hd + 255) / 256, 256, 0, stream>>>(ws_locvec, Bd * Hd + Hd);
    // 2) usr_vec matvec
    strnn_usr_kernel<<<2, 256, 0, stream>>>(Wih, h0, ws_usr);
    // 3) fused double-GEMM with WMMA f32 16x16x4
    strnn_main_kernel<<<Bd * (Ld / 16), 128, 0, stream>>>(
        tdu, tdl, ldu, ldl, cur_loc, loc_len, Wtu, Wtl, Wsu, Wsl, locw, ws_locvec);
    // 4) softmax
    strnn_softmax_kernel<<<Bd, 256, 0, stream>>>(ws_locvec, ws_usr, out);
}